// NLSPN_44899588112818
// MI455X (gfx1250) — compile-verified
//
#include <hip/hip_runtime.h>
#include <stdint.h>

#define DEV __device__ __forceinline__

constexpr int Bc = 4, Hc = 480, Wc = 640;
constexpr int HW   = Hc * Wc;          // 307200
constexpr int NPIX = Bc * HW;          // 1228800
constexpr int TILE = 256;              // 8 wave32 per block
constexpr int NBLK = NPIX / TILE;      // 4800 (exact)
constexpr int PAD  = Wc + 1;           // halo so idx, idx+1, idx+W, idx+W+1 stay in-buffer
constexpr int CHUNK_BYTES = 28672;     // 7 * 4096 (9 taps * 3 words * 256 px = 27648, padded)
constexpr int CHUNK_WORDS = CHUNK_BYTES / 4;
constexpr int PROP_TIME = 18;

// 16-byte gcc-style int vector, matching the builtin's parameter type
typedef int v4i_t __attribute__((vector_size(16)));
typedef __attribute__((address_space(1))) v4i_t* gptr_v4i;
typedef __attribute__((address_space(3))) v4i_t* lptr_v4i;

// ---------- CDNA5 async global->LDS bulk copy (ASYNCcnt engine) ----------
DEV void async_g2l_b128(const void* gsrc, void* ldst) {
#if __has_builtin(__builtin_amdgcn_global_load_async_to_lds_b128)
  __builtin_amdgcn_global_load_async_to_lds_b128(
      (gptr_v4i)gsrc, (lptr_v4i)ldst, 0, 0);
#else
  unsigned l32 =
      (unsigned)(unsigned long long)(__attribute__((address_space(3))) char*)ldst;
  unsigned long long g64 = (unsigned long long)gsrc;
  asm volatile("global_load_async_to_lds_b128 %0, %1, off"
               :: "v"(l32), "v"(g64) : "memory");
#endif
}
DEV void wait_async0() {
#if __has_builtin(__builtin_amdgcn_s_wait_asynccnt)
  __builtin_amdgcn_s_wait_asynccnt(0);
#else
  asm volatile("s_wait_asynccnt 0x0" ::: "memory");
#endif
}

// ---------- fp16 pack/unpack ----------
DEV unsigned pack_h2(float a, float b) {
  _Float16 ha = (_Float16)a, hb = (_Float16)b;
  unsigned short ua, ub;
  __builtin_memcpy(&ua, &ha, 2);
  __builtin_memcpy(&ub, &hb, 2);
  return (unsigned)ua | ((unsigned)ub << 16);
}
DEV float h_lo(unsigned u) {
  unsigned short s = (unsigned short)u;
  _Float16 h;
  __builtin_memcpy(&h, &s, 2);
  return (float)h;
}
DEV float h_hi(unsigned u) { return h_lo(u >> 16); }

// ---------- prep: fold offsets/affinity/confidence/mask into a gather table ----------
// chunk layout per block (bi): u32 chunk[9 taps][3 fields][256 px]
//   field 0: padded base index (b*HW + y0*W + x0 + PAD), field1: half2(w00,w01), field2: half2(w10,w11)
__global__ __launch_bounds__(TILE) void nlspn_prep(
    const float* __restrict__ feat_init, const float* __restrict__ guidance,
    const float* __restrict__ confidence, const float* __restrict__ feat_fix,
    const float* __restrict__ aff_scale, unsigned* __restrict__ table,
    float* __restrict__ fbufA, float* __restrict__ fbufB, float* __restrict__ cbuf) {
  const int t = threadIdx.x, bi = blockIdx.x;
  const int p = bi * TILE + t;

  // zero the halo pads of both ping-pong buffers (so 0-weight * pad is exactly 0)
  {
    const int j = bi * TILE + t;
    if (j < PAD) {
      fbufA[j] = 0.f; fbufA[PAD + NPIX + j] = 0.f;
      fbufB[j] = 0.f; fbufB[PAD + NPIX + j] = 0.f;
    }
  }

  const int b = p / HW, r = p % HW;
  const int y = r / Wc, x = r % Wc;

  const float* gb = guidance + (size_t)b * 24 * HW + r;
  float o[16];
#pragma unroll
  for (int c = 0; c < 16; ++c) o[c] = gb[(size_t)c * HW];

  const float scale = aff_scale[0] + 1e-8f;
  float a[8], s = 1e-4f;
#pragma unroll
  for (int c = 0; c < 8; ++c) {
    a[c] = tanhf(gb[(size_t)(16 + c) * HW]) / scale;
    s += fabsf(a[c]);
  }
  s = fmaxf(s, 1.0f);
  float suma = 0.f;
#pragma unroll
  for (int c = 0; c < 8; ++c) { a[c] /= s; suma += a[c]; }
  const float aref = 1.0f - suma;

  const float fix   = feat_fix[p];
  const float m     = fix > 0.f ? 1.f : 0.f;
  const float one_m = 1.f - m;
  fbufA[PAD + p] = one_m * feat_init[p] + m * fix;   // f0
  cbuf[p]        = m * fix;                          // loop-invariant additive term

  unsigned* chunk = table + (size_t)bi * CHUNK_WORDS;
  const int kyv[9] = {-1,-1,-1, 0,0,0, 1,1,1};
  const int kxv[9] = {-1, 0, 1,-1,0,1,-1,0,1};
#pragma unroll
  for (int k = 0; k < 9; ++k) {
    float dy, dx, ak;
    if (k < 4)       { dy = o[2*k];   dx = o[2*k+1]; ak = a[k];   }
    else if (k == 4) { dy = 0.f;      dx = 0.f;      ak = aref;   }
    else             { dy = o[2*k-2]; dx = o[2*k-1]; ak = a[k-1]; }
    const float py = (float)(y + kyv[k]) + dy;
    const float px = (float)(x + kxv[k]) + dx;
    const float fy = floorf(py), fx = floorf(px);
    const int y0 = (int)fy, x0 = (int)fx;
    const float wy = py - fy, wx = px - fx;
    const float bw[4] = {(1.f-wy)*(1.f-wx), (1.f-wy)*wx, wy*(1.f-wx), wy*wx};
    const bool safe = (y0 >= -1) && (y0 <= Hc - 1) && (x0 >= -1) && (x0 <= Wc - 1);
    const int base = safe ? (b * HW + y0 * Wc + x0 + PAD) : PAD;
    const float cw = ak * one_m;
    float ww[4];
    const int dyc[4] = {0, 0, 1, 1}, dxc[4] = {0, 1, 0, 1};
#pragma unroll
    for (int c = 0; c < 4; ++c) {
      const int yy = y0 + dyc[c], xx = x0 + dxc[c];
      const bool v = (yy >= 0) && (yy < Hc) && (xx >= 0) && (xx < Wc);
      float wv = 0.f;
      if (v) {
        const int q = b * HW + yy * Wc + xx;
        const float ce = (feat_fix[q] > 0.f) ? 1.f : confidence[q];  // folded conf
        wv = bw[c] * cw * ce;
      }
      ww[c] = wv;
    }
    chunk[(k * 3 + 0) * TILE + t] = (unsigned)base;
    chunk[(k * 3 + 1) * TILE + t] = pack_h2(ww[0], ww[1]);
    chunk[(k * 3 + 2) * TILE + t] = pack_h2(ww[2], ww[3]);
  }
}

// ---------- one propagation step: async-stage table chunk to LDS, gather-FMA ----------
__global__ __launch_bounds__(TILE) void nlspn_iter(
    const unsigned* __restrict__ table, const float* __restrict__ fsrc,
    const float* __restrict__ cbuf, float* __restrict__ fdst) {
  __shared__ __align__(16) unsigned lds[CHUNK_WORDS];
  const int t = threadIdx.x, bi = blockIdx.x;
  const char* g = (const char*)(table + (size_t)bi * CHUNK_WORDS);
  char* l = (char*)lds;
#pragma unroll
  for (int i = 0; i < 7; ++i)   // 7 x (256 lanes * 16B) = 28 KB, fully coalesced
    async_g2l_b128(g + i * 4096 + t * 16, l + i * 4096 + t * 16);

  const int p = bi * TILE + t;
  float acc = cbuf[p];          // overlap with async DMA

  wait_async0();
  __syncthreads();

#pragma unroll
  for (int k = 0; k < 9; ++k) {
    const unsigned base = lds[(k * 3 + 0) * TILE + t];
    const unsigned w01  = lds[(k * 3 + 1) * TILE + t];
    const unsigned w23  = lds[(k * 3 + 2) * TILE + t];
    const float* sp = fsrc + base;
    const float f00 = sp[0], f01 = sp[1], f10 = sp[Wc], f11 = sp[Wc + 1];
    acc = fmaf(h_lo(w01), f00, acc);
    acc = fmaf(h_hi(w01), f01, acc);
    acc = fmaf(h_lo(w23), f10, acc);
    acc = fmaf(h_hi(w23), f11, acc);
  }
  fdst[p] = acc;
}

extern "C" void kernel_launch(void* const* d_in, const int* in_sizes, int n_in,
                              void* d_out, int out_size, void* d_ws, size_t ws_size,
                              hipStream_t stream) {
  const float* feat_init  = (const float*)d_in[0];
  const float* guidance   = (const float*)d_in[1];
  const float* confidence = (const float*)d_in[2];
  const float* feat_fix   = (const float*)d_in[3];
  const float* aff_scale  = (const float*)d_in[4];
  float* out = (float*)d_out;

  char* ws = (char*)d_ws;
  const size_t fbuf_bytes = (((size_t)(NPIX + 2 * PAD) * 4) + 255) & ~(size_t)255;
  const size_t cbuf_bytes = (((size_t)NPIX * 4) + 255) & ~(size_t)255;
  float*    fbufA = (float*)(ws);
  float*    fbufB = (float*)(ws + fbuf_bytes);
  float*    cbuf  = (float*)(ws + 2 * fbuf_bytes);
  unsigned* table = (unsigned*)(ws + 2 * fbuf_bytes + cbuf_bytes);
  // total ws use: ~2*4.9MB + 4.9MB + 137.6MB ~= 152 MB

  nlspn_prep<<<NBLK, TILE, 0, stream>>>(feat_init, guidance, confidence, feat_fix,
                                        aff_scale, table, fbufA, fbufB, cbuf);

  float* bufs[2] = {fbufA, fbufB};
  for (int i = 0; i < PROP_TIME; ++i) {
    const float* src = bufs[i & 1];
    float* dst = (i == PROP_TIME - 1) ? out : (bufs[(i + 1) & 1] + PAD);
    nlspn_iter<<<NBLK, TILE, 0, stream>>>(table, src, cbuf, dst);
  }
  (void)in_sizes; (void)n_in; (void)out_size; (void)ws_size;
}